// GNNModel_2637109920399
// MI455X (gfx1250) — compile-verified
//
#include <hip/hip_runtime.h>
#include <stdint.h>

typedef float v2f __attribute__((ext_vector_type(2)));
typedef float v8f __attribute__((ext_vector_type(8)));

#define N_NODES 100000
#define N_EDGES 1200000
#define FIN_CH  128
#define HID     64

// ---------------- degree / normalization ----------------
__global__ __launch_bounds__(256) void k_deg_init(float* __restrict__ deg) {
    int i = blockIdx.x * 256 + threadIdx.x;
    if (i < N_NODES) deg[i] = 1.0f;              // self-loop contribution
}

__global__ __launch_bounds__(256) void k_deg_accum(const long long* __restrict__ ei,
                                                   float* __restrict__ deg) {
    int e = blockIdx.x * 256 + threadIdx.x;
    if (e < N_EDGES) atomicAdd(&deg[(int)ei[N_EDGES + e]], 1.0f);   // dst = edge_index[1]
}

__global__ __launch_bounds__(256) void k_deg_rsqrt(float* __restrict__ deg) {
    int i = blockIdx.x * 256 + threadIdx.x;
    if (i < N_NODES) deg[i] = rsqrtf(deg[i]);
}

// ---------------- fp32 WMMA GEMM:  T[N,64] = A[N,K] @ W[K,64] ----------------
// One wave -> 16x16 output tile. Block = 4 waves -> 16 rows x 64 cols.
template <int K>
__global__ __launch_bounds__(128) void k_gemm_wmma_f32(const float* __restrict__ A,
                                                       const float* __restrict__ W,
                                                       float* __restrict__ T) {
    const int lane    = threadIdx.x & 31;
    const int wid     = threadIdx.x >> 5;
    const int rowbase = blockIdx.x * 16;          // N divisible by 16
    const int m       = lane & 15;
    const int khalf   = (lane >> 4) * 2;          // lanes 0-15: K=0,1 ; lanes 16-31: K=2,3
    const int col     = wid * 16 + (lane & 15);   // B/C/D: N striped across lanes

    v8f acc = {};
    const float* __restrict__ arow = A + (size_t)(rowbase + m) * K + khalf;
#pragma unroll 4
    for (int k0 = 0; k0 < K; k0 += 4) {
        v2f a = *(const v2f*)(arow + k0);         // A[m][k0+khalf .. +1], 8B aligned
        v2f b;
        b.x = W[(k0 + khalf) * HID + col];
        b.y = W[(k0 + khalf + 1) * HID + col];
        // (neg_a, A, neg_b, B, c_mod, C, reuse_a, reuse_b)
        acc = __builtin_amdgcn_wmma_f32_16x16x4_f32(false, a, false, b,
                                                    (short)0, acc, false, false);
    }
    // C/D layout: VGPR i -> M = i + 8*(lane>=16), N = lane&15
    const int rowoff = (lane >> 4) * 8;
#pragma unroll
    for (int i = 0; i < 8; ++i)
        T[(size_t)(rowbase + rowoff + i) * HID + col] = acc[i];
}

// ---------------- aggregation, 64-wide ----------------
// out[n][c] = b[c] + t[n][c] * dinv[n]^2   (bias + self-loop term)
__global__ __launch_bounds__(256) void k_agg_init64(const float* __restrict__ t,
                                                    const float* __restrict__ dinv,
                                                    const float* __restrict__ bias,
                                                    float* __restrict__ out) {
    int i = blockIdx.x * 256 + threadIdx.x;       // over N*64, exact multiple
    int n = i >> 6, c = i & 63;
    float di = dinv[n];
    out[i] = bias[c] + t[i] * di * di;
}

// wave per edge: 32 lanes x 2 channels, coalesced gather + f32 atomic scatter
__global__ __launch_bounds__(256) void k_agg_edges64(const float* __restrict__ t,
                                                     const float* __restrict__ dinv,
                                                     const long long* __restrict__ ei,
                                                     float* __restrict__ out) {
    int lane = threadIdx.x & 31;
    int e    = blockIdx.x * 8 + (threadIdx.x >> 5);   // E divisible by 8
    int s = (int)ei[e];
    int d = (int)ei[N_EDGES + e];
    float w = dinv[s] * dinv[d];
    v2f v = *(const v2f*)(t + (size_t)s * HID + lane * 2);
    float* o = out + (size_t)d * HID + lane * 2;
    atomicAdd(o,     v.x * w);
    atomicAdd(o + 1, v.y * w);
}

__global__ __launch_bounds__(256) void k_relu(float* __restrict__ x) {
    int i = blockIdx.x * 256 + threadIdx.x;       // over N*64, exact multiple
    float v = x[i];
    x[i] = v > 0.0f ? v : 0.0f;
}

// ---------------- layer 2: matvec 64 -> 1 ----------------
__global__ __launch_bounds__(256) void k_matvec64(const float* __restrict__ h,
                                                  const float* __restrict__ w,
                                                  float* __restrict__ t2) {
    int lane = threadIdx.x & 31;
    int n    = blockIdx.x * 8 + (threadIdx.x >> 5);   // N divisible by 8
    const float* hr = h + (size_t)n * HID;
    float p = hr[lane] * w[lane] + hr[lane + 32] * w[lane + 32];
#pragma unroll
    for (int off = 16; off > 0; off >>= 1) p += __shfl_xor(p, off, 32);
    if (lane == 0) t2[n] = p;
}

__global__ __launch_bounds__(256) void k_agg_init1(const float* __restrict__ t2,
                                                   const float* __restrict__ dinv,
                                                   const float* __restrict__ b2,
                                                   float* __restrict__ out) {
    int n = blockIdx.x * 256 + threadIdx.x;
    if (n < N_NODES) {
        float di = dinv[n];
        out[n] = b2[0] + t2[n] * di * di;
    }
}

__global__ __launch_bounds__(256) void k_agg_edges1(const float* __restrict__ t2,
                                                    const float* __restrict__ dinv,
                                                    const long long* __restrict__ ei,
                                                    float* __restrict__ out) {
    int e = blockIdx.x * 256 + threadIdx.x;
    if (e < N_EDGES) {
        int s = (int)ei[e];
        int d = (int)ei[N_EDGES + e];
        atomicAdd(&out[d], t2[s] * dinv[s] * dinv[d]);
    }
}

// ---------------- MLP head: y = relu(h*Wm1 + bm1) @ Wm2 + bm2 ----------------
__global__ __launch_bounds__(256) void k_mlp_head(const float* __restrict__ h3,
                                                  const float* __restrict__ Wm1,
                                                  const float* __restrict__ bm1,
                                                  const float* __restrict__ Wm2,
                                                  const float* __restrict__ bm2,
                                                  float* __restrict__ y) {
    __shared__ float w1[HID], bb[HID], w2[HID];
    if (threadIdx.x < HID) {
        w1[threadIdx.x] = Wm1[threadIdx.x];
        bb[threadIdx.x] = bm1[threadIdx.x];
        w2[threadIdx.x] = Wm2[threadIdx.x];
    }
    __syncthreads();
    int n = blockIdx.x * 256 + threadIdx.x;
    if (n >= N_NODES) return;
    float s   = h3[n];
    float acc = bm2[0];
#pragma unroll 8
    for (int c = 0; c < HID; ++c)
        acc += fmaxf(fmaf(s, w1[c], bb[c]), 0.0f) * w2[c];
    y[n] = acc;
}

// ---------------- host launcher ----------------
extern "C" void kernel_launch(void* const* d_in, const int* in_sizes, int n_in,
                              void* d_out, int out_size, void* d_ws, size_t ws_size,
                              hipStream_t stream) {
    const float*     x   = (const float*)d_in[0];
    const long long* ei  = (const long long*)d_in[1];   // int64 per reference
    const float*     W0  = (const float*)d_in[2];
    const float*     b0  = (const float*)d_in[3];
    const float*     W1  = (const float*)d_in[4];
    const float*     b1  = (const float*)d_in[5];
    const float*     W2  = (const float*)d_in[6];
    const float*     b2  = (const float*)d_in[7];
    const float*     Wm1 = (const float*)d_in[8];
    const float*     bm1 = (const float*)d_in[9];
    const float*     Wm2 = (const float*)d_in[10];
    const float*     bm2 = (const float*)d_in[11];
    float* y = (float*)d_out;

    // workspace layout (floats): dinv[N] | bufA[N*64] | bufB[N*64] | bufC[N] | bufD[N]
    float* dinv = (float*)d_ws;
    float* bufA = dinv + N_NODES;
    float* bufB = bufA + (size_t)N_NODES * HID;
    float* bufC = bufB + (size_t)N_NODES * HID;
    float* bufD = bufC + N_NODES;

    const int gN   = (N_NODES + 255) / 256;       // 391
    const int gE   = (N_EDGES + 255) / 256;       // 4688
    const int gNH  = N_NODES * HID / 256;         // 25000 (exact)
    const int gRow = N_NODES / 16;                // 6250  (exact)
    const int gEw  = N_EDGES / 8;                 // 150000 (exact)
    const int gNw  = N_NODES / 8;                 // 12500 (exact)

    // normalization
    k_deg_init  <<<gN, 256, 0, stream>>>(dinv);
    k_deg_accum <<<gE, 256, 0, stream>>>(ei, dinv);
    k_deg_rsqrt <<<gN, 256, 0, stream>>>(dinv);

    // layer 0: t0 = x @ W0 ; h1 = relu(scatter(t0) + b0)
    k_gemm_wmma_f32<FIN_CH><<<gRow, 128, 0, stream>>>(x, W0, bufA);
    k_agg_init64 <<<gNH, 256, 0, stream>>>(bufA, dinv, b0, bufB);
    k_agg_edges64<<<gEw, 256, 0, stream>>>(bufA, dinv, ei, bufB);
    k_relu       <<<gNH, 256, 0, stream>>>(bufB);

    // layer 1: t1 = h1 @ W1 ; h2 = relu(scatter(t1) + b1)
    k_gemm_wmma_f32<HID><<<gRow, 128, 0, stream>>>(bufB, W1, bufA);
    k_agg_init64 <<<gNH, 256, 0, stream>>>(bufA, dinv, b1, bufB);
    k_agg_edges64<<<gEw, 256, 0, stream>>>(bufA, dinv, ei, bufB);
    k_relu       <<<gNH, 256, 0, stream>>>(bufB);

    // layer 2: t2 = h2 @ W2 (64->1) ; h3 = scatter(t2) + b2  (no relu)
    k_matvec64   <<<gNw, 256, 0, stream>>>(bufB, W2, bufC);
    k_agg_init1  <<<gN, 256, 0, stream>>>(bufC, dinv, b2, bufD);
    k_agg_edges1 <<<gE, 256, 0, stream>>>(bufC, dinv, ei, bufD);

    // MLP head
    k_mlp_head   <<<gN, 256, 0, stream>>>(bufD, Wm1, bm1, Wm2, bm2, y);
}